// ItemCodeLayer_39779987096003
// MI455X (gfx1250) — compile-verified
//
#include <hip/hip_runtime.h>
#include <cstdint>

// PQ embedding gather for MI455X (gfx1250, wave32).
//   input_ids  : [B*S]            int32
//   item_codes : [NUM_ITEMS+2, 8] int32 (byte codes 0..255)
//   centroids  : [8, 256, 16]     f32   (128 KB total)
//   out        : [B*S, 128]       f32   (100 MB -> store-bandwidth bound)
//
// Strategy: stage the whole centroid table in LDS via CDNA5 async
// global->LDS b128 copies (ASYNCcnt), gather sub-vectors with ds_load_b128,
// emit one coalesced 512B non-temporal b128 store line per wave per row.

#define PQ_M        8
#define VALS        256
#define SUB_DIM     16
#define ROW_FLOATS  (PQ_M * SUB_DIM)              // 128
#define CENT_FLOATS (PQ_M * VALS * SUB_DIM)       // 32768
#define CENT_BYTES  (CENT_FLOATS * 4)             // 131072 (fits 320KB LDS/WGP)

typedef float v4f __attribute__((ext_vector_type(4)));

extern "C" __global__ __launch_bounds__(256)
void pq_gather_kernel(const int*   __restrict__ input_ids,
                      const int*   __restrict__ item_codes,
                      const float* __restrict__ centroids,
                      float*       __restrict__ out,
                      int nrows)
{
    extern __shared__ __align__(16) float sh[];   // 128 KB centroid mirror

    const unsigned t = threadIdx.x;

    // ---- Async-fill LDS with the centroid table (8192 x b128 chunks).
    // 256 threads x 32 chunks; each lane copies 16B memory->LDS with no VGPR
    // round trip. Tracked by ASYNCcnt.
    {
        const unsigned           lds_base = (unsigned)(uintptr_t)(void*)sh;
        const unsigned long long gbase    = (unsigned long long)(uintptr_t)centroids;
        #pragma unroll
        for (int i = 0; i < 32; ++i) {
            const unsigned off = (t + (unsigned)i * 256u) * 16u;
            const unsigned           lds_off = lds_base + off;
            const unsigned long long gaddr   = gbase + off;
            asm volatile("global_load_async_to_lds_b128 %0, %1, off"
                         :: "v"(lds_off), "v"(gaddr) : "memory");
        }
        asm volatile("s_wait_asynccnt 0" ::: "memory");
    }
    __syncthreads();

    // ---- One wave32 per output row: lane L -> pq dim (L>>2), floats [(L&3)*4, +4).
    const unsigned lane  = t & 31u;
    const unsigned wave  = t >> 5u;
    const unsigned gwave = blockIdx.x * (blockDim.x >> 5u) + wave;
    const unsigned nwave = gridDim.x * (blockDim.x >> 5u);
    const unsigned m     = lane >> 2u;          // PQ dim 0..7
    const unsigned sub   = (lane & 3u) << 2u;   // float offset inside sub-vector

    for (unsigned row = gwave; row < (unsigned)nrows; row += nwave) {
        // row is wave-uniform; force scalar so the id load can use SMEM path.
        const int r  = __builtin_amdgcn_readfirstlane((int)row);
        const int id = input_ids[r];
        // 8 distinct dwords per wave (each read by 4 lanes) from the L2-resident
        // 32MB code table.
        const unsigned code = (unsigned)item_codes[(size_t)(unsigned)id * PQ_M + m];
        // LDS gather: ds_load_b128, 16B per lane, conflict-friendly stride.
        const v4f v = *reinterpret_cast<const v4f*>(
            &sh[((m << 8) + code) * SUB_DIM + sub]);
        // Coalesced 512B/wave store; non-temporal: output is write-once, keep
        // it from evicting the code table out of L2.
        v4f* dst = reinterpret_cast<v4f*>(out + (size_t)r * ROW_FLOATS + lane * 4u);
        __builtin_nontemporal_store(v, dst);
    }
}

extern "C" void kernel_launch(void* const* d_in, const int* in_sizes, int n_in,
                              void* d_out, int out_size, void* d_ws, size_t ws_size,
                              hipStream_t stream)
{
    (void)n_in; (void)out_size; (void)d_ws; (void)ws_size;
    const int*   input_ids  = (const int*)d_in[0];
    const int*   item_codes = (const int*)d_in[1];
    const float* centroids  = (const float*)d_in[2];
    float*       out        = (float*)d_out;

    const int nrows = in_sizes[0];   // B*S = 204800

    // Persistent grid: 1024 blocks x 8 waves; 128KB dynamic LDS -> 2 WGs/WGP.
    pq_gather_kernel<<<1024, 256, CENT_BYTES, stream>>>(
        input_ids, item_codes, centroids, out, nrows);
}